// MultiModelMLP_14723147890777
// MI455X (gfx1250) — compile-verified
//
#include <hip/hip_runtime.h>

// ---------------------------------------------------------------------------
// Multi-model MLP (64 models, 6->64, 3x(64->64), 64->3, LN+ReLU) for gfx1250.
//
// Strategy:
//   1) Bin samples by model id (histogram -> scan -> scatter perm array in ws)
//   2) One block per (model, slice). Model weights staged in LDS (~55 KB).
//      Each wave processes 16-sample tiles of its model with
//      V_WMMA_F32_16X16X4_F32 (exact fp32, matches reference precision).
//   3) Activations live in registers in the WMMA C/D layout; the B-fragment
//      reshuffle between layers is a lane-xor-16 exchange (wave32).
// ---------------------------------------------------------------------------

#define NMODEL 64
#define IN_F   6
#define HID    64
#define OUT_F  3
#define LN_EPS 1e-5f
#define BPM    16   // blocks per model

typedef float v2f __attribute__((ext_vector_type(2)));
typedef float v8f __attribute__((ext_vector_type(8)));

// LDS layout (float offsets)
#define LW_HID 0        // W1,W2,W3: 3*4096
#define LW0    12288    // W0 padded to 64x8 (K padded 6->8 with zeros)
#define LW4    12800    // W4: 3*64
#define LBIAS  12992    // B0..B3: 4*64
#define LGAIN  13248    // G0..G3: 4*64
#define LSHFT  13504    // H0..H3: 4*64
#define LB4    13760    // B4 (padded to 4)
#define LDS_FLOATS 13764

__device__ __forceinline__ v8f wmma_f32(v2f a, v2f b, v8f c) {
  // D = A(16x4) * B(4x16) + C, full fp32
  return __builtin_amdgcn_wmma_f32_16x16x4_f32(
      /*neg_a=*/false, a, /*neg_b=*/false, b,
      /*c_mod=*/(short)0, c, /*reuse_a=*/false, /*reuse_b=*/false);
}

// Bias + LayerNorm + ReLU on a wave-held 64x16 tile in D layout.
// lane(n,hi) holds acc[t][v] = y[16t+8hi+v][n].
__device__ __forceinline__ void ln_relu(v8f acc[4], float act[32],
                                        const float* lds, int li, int hi) {
  float sum = 0.f, sq = 0.f;
#pragma unroll
  for (int t = 0; t < 4; ++t)
#pragma unroll
    for (int v = 0; v < 8; ++v) {
      int mrow = 16 * t + 8 * hi + v;
      float z = acc[t][v] + lds[LBIAS + li * 64 + mrow];
      acc[t][v] = z;
      sum += z;
      sq  += z * z;
    }
  // combine the two half-columns (lanes n and n+16)
  sum += __shfl_xor(sum, 16, 32);
  sq  += __shfl_xor(sq, 16, 32);
  float mu  = sum * (1.0f / 64.0f);
  float var = sq * (1.0f / 64.0f) - mu * mu;
  float rs  = rsqrtf(var + LN_EPS);
#pragma unroll
  for (int t = 0; t < 4; ++t)
#pragma unroll
    for (int v = 0; v < 8; ++v) {
      int mrow = 16 * t + 8 * hi + v;
      float z = (acc[t][v] - mu) * rs;
      z = z * lds[LGAIN + li * 64 + mrow] + lds[LSHFT + li * 64 + mrow];
      act[t * 8 + v] = fmaxf(z, 0.f);
    }
}

// ---------------- binning pre-pass ----------------

__global__ void zero_k(int* p) {
  int i = blockIdx.x * blockDim.x + threadIdx.x;
  if (i < 128) p[i] = 0;  // counts[64] + cursors[64]
}

__global__ void count_k(const int* __restrict__ idx, int* __restrict__ cnt, int n) {
  int i = blockIdx.x * blockDim.x + threadIdx.x;
  if (i < n) atomicAdd(&cnt[idx[i]], 1);
}

__global__ void scan_k(const int* __restrict__ cnt, int* __restrict__ off) {
  if (threadIdx.x == 0) {
    int a = 0;
    for (int i = 0; i < NMODEL; ++i) { off[i] = a; a += cnt[i]; }
    off[NMODEL] = a;
  }
}

__global__ void scatter_k(const int* __restrict__ idx, const int* __restrict__ off,
                          int* __restrict__ cur, int* __restrict__ perm, int n) {
  int i = blockIdx.x * blockDim.x + threadIdx.x;
  if (i < n) {
    int m = idx[i];
    int p = off[m] + atomicAdd(&cur[m], 1);
    perm[p] = i;
  }
}

// ---------------- main MLP kernel ----------------

__global__ __launch_bounds__(256) void mlp_k(
    const float* __restrict__ x, const int* __restrict__ perm,
    const int* __restrict__ off,
    const float* __restrict__ W0, const float* __restrict__ B0,
    const float* __restrict__ G0, const float* __restrict__ H0,
    const float* __restrict__ W1, const float* __restrict__ B1,
    const float* __restrict__ G1, const float* __restrict__ H1,
    const float* __restrict__ W2, const float* __restrict__ B2,
    const float* __restrict__ G2, const float* __restrict__ H2,
    const float* __restrict__ W3, const float* __restrict__ B3,
    const float* __restrict__ G3, const float* __restrict__ H3,
    const float* __restrict__ W4, const float* __restrict__ B4,
    float* __restrict__ out) {
  __shared__ float lds[LDS_FLOATS];
  const int tid   = threadIdx.x;
  const int m     = blockIdx.x / BPM;
  const int slice = blockIdx.x % BPM;

  // ---- cooperative weight staging into LDS ----
  {
    const float* w1 = W1 + m * 4096;
    const float* w2 = W2 + m * 4096;
    const float* w3 = W3 + m * 4096;
    for (int i = tid; i < 4096; i += 256) {
      lds[LW_HID + i]        = w1[i];
      lds[LW_HID + 4096 + i] = w2[i];
      lds[LW_HID + 8192 + i] = w3[i];
    }
    for (int i = tid; i < 512; i += 256) {  // W0 padded 64x8
      int r = i >> 3, c = i & 7;
      lds[LW0 + i] = (c < IN_F) ? W0[m * HID * IN_F + r * IN_F + c] : 0.0f;
    }
    if (tid < OUT_F * HID) lds[LW4 + tid] = W4[m * OUT_F * HID + tid];
    if (tid < HID) {
      lds[LBIAS + 0 * 64 + tid] = B0[m * 64 + tid];
      lds[LBIAS + 1 * 64 + tid] = B1[m * 64 + tid];
      lds[LBIAS + 2 * 64 + tid] = B2[m * 64 + tid];
      lds[LBIAS + 3 * 64 + tid] = B3[m * 64 + tid];
      lds[LGAIN + 0 * 64 + tid] = G0[m * 64 + tid];
      lds[LGAIN + 1 * 64 + tid] = G1[m * 64 + tid];
      lds[LGAIN + 2 * 64 + tid] = G2[m * 64 + tid];
      lds[LGAIN + 3 * 64 + tid] = G3[m * 64 + tid];
      lds[LSHFT + 0 * 64 + tid] = H0[m * 64 + tid];
      lds[LSHFT + 1 * 64 + tid] = H1[m * 64 + tid];
      lds[LSHFT + 2 * 64 + tid] = H2[m * 64 + tid];
      lds[LSHFT + 3 * 64 + tid] = H3[m * 64 + tid];
    }
    if (tid < 4) lds[LB4 + tid] = (tid < OUT_F) ? B4[m * OUT_F + tid] : 0.f;
  }
  __syncthreads();

  const int seg0 = off[m], seg1 = off[m + 1];
  const int ntiles = (seg1 - seg0 + 15) >> 4;
  const int wave = tid >> 5;
  const int lane = tid & 31;
  const int nl   = lane & 15;  // sample column within tile / A-row within tile
  const int hi   = lane >> 4;  // lane half (selects K pair / row-half)

  for (int tile = slice * 8 + wave; tile < ntiles; tile += BPM * 8) {
    int  pos   = seg0 + tile * 16 + nl;
    bool valid = pos < seg1;
    int  s     = perm[valid ? pos : (seg1 - 1)];

    v8f acc[4];
#pragma unroll
    for (int t = 0; t < 4; ++t)
#pragma unroll
      for (int v = 0; v < 8; ++v) acc[t][v] = 0.0f;

    // ---- layer 0: 6 -> 64 as two K=4 WMMA steps (K padded to 8) ----
    {
      const float* xs = x + (long)s * IN_F;
      float e0 = xs[2 * hi], e1 = xs[2 * hi + 1];
      float e4 = xs[4], e5 = xs[5];
      v2f b0; b0.x = e0;                b0.y = e1;
      v2f b1; b1.x = hi ? 0.f : e4;     b1.y = hi ? 0.f : e5;
#pragma unroll
      for (int t = 0; t < 4; ++t) {
        const float* wrow = &lds[LW0 + (16 * t + nl) * 8];
        v2f a0; a0.x = wrow[2 * hi];     a0.y = wrow[2 * hi + 1];
        acc[t] = wmma_f32(a0, b0, acc[t]);
        v2f a1; a1.x = wrow[4 + 2 * hi]; a1.y = wrow[4 + 2 * hi + 1];
        acc[t] = wmma_f32(a1, b1, acc[t]);
      }
    }
    float act[32];
    ln_relu(acc, act, lds, 0, hi);

    // ---- hidden layers 1..3: 64x64 GEMM = 4 M-tiles x 16 K-steps ----
#pragma unroll 1
    for (int li = 1; li <= 3; ++li) {
      const float* wl = &lds[LW_HID + (li - 1) * 4096];
      v8f nacc[4];
#pragma unroll
      for (int t = 0; t < 4; ++t)
#pragma unroll
        for (int v = 0; v < 8; ++v) nacc[t][v] = 0.0f;

#pragma unroll
      for (int j = 0; j < 16; ++j) {
        // Build B fragment (H[4j+2hi..+1][n]) from register act[] (D layout).
        // The 4 K-values of this group live in 4 compile-time slots; the half
        // not held locally is fetched from the xor-16 partner lane.
        float a0 = act[(((4 * j + 0) >> 4) * 8) + ((4 * j + 0) & 7)];
        float a1 = act[(((4 * j + 1) >> 4) * 8) + ((4 * j + 1) & 7)];
        float a2 = act[(((4 * j + 2) >> 4) * 8) + ((4 * j + 2) & 7)];
        float a3 = act[(((4 * j + 3) >> 4) * 8) + ((4 * j + 3) & 7)];
        float sx = hi ? a0 : a2;  // value our partner needs
        float sy = hi ? a1 : a3;
        float rx = __shfl_xor(sx, 16, 32);
        float ry = __shfl_xor(sy, 16, 32);
        bool loc = (((j >> 1) & 1) == hi);  // our pair is held locally?
        float lx = hi ? a2 : a0;
        float ly = hi ? a3 : a1;
        v2f bf; bf.x = loc ? lx : rx; bf.y = loc ? ly : ry;
#pragma unroll
        for (int t = 0; t < 4; ++t) {
          const float* wr = &wl[(16 * t + nl) * 64 + 4 * j + 2 * hi];
          v2f af; af.x = wr[0]; af.y = wr[1];
          nacc[t] = wmma_f32(af, bf, nacc[t]);
        }
      }
      ln_relu(nacc, act, lds, li, hi);
    }

    // ---- final layer 64 -> 3 (VALU) ----
    float p0 = 0.f, p1 = 0.f, p2 = 0.f;
#pragma unroll
    for (int t = 0; t < 4; ++t)
#pragma unroll
      for (int v = 0; v < 8; ++v) {
        int k = 16 * t + 8 * hi + v;
        float h = act[t * 8 + v];
        p0 += h * lds[LW4 + 0 * 64 + k];
        p1 += h * lds[LW4 + 1 * 64 + k];
        p2 += h * lds[LW4 + 2 * 64 + k];
      }
    p0 += __shfl_xor(p0, 16, 32);
    p1 += __shfl_xor(p1, 16, 32);
    p2 += __shfl_xor(p2, 16, 32);
    if (hi == 0 && valid) {
      out[(long)s * 3 + 0] = p0 + lds[LB4 + 0];
      out[(long)s * 3 + 1] = p1 + lds[LB4 + 1];
      out[(long)s * 3 + 2] = p2 + lds[LB4 + 2];
    }
  }
}

// ---------------- launch ----------------

extern "C" void kernel_launch(void* const* d_in, const int* in_sizes, int n_in,
                              void* d_out, int out_size, void* d_ws, size_t ws_size,
                              hipStream_t stream) {
  const float* x   = (const float*)d_in[0];
  const int* idx   = (const int*)d_in[1];
  const float* W0  = (const float*)d_in[2];
  const float* B0  = (const float*)d_in[3];
  const float* G0  = (const float*)d_in[4];
  const float* H0  = (const float*)d_in[5];
  const float* W1  = (const float*)d_in[6];
  const float* B1  = (const float*)d_in[7];
  const float* G1  = (const float*)d_in[8];
  const float* H1  = (const float*)d_in[9];
  const float* W2  = (const float*)d_in[10];
  const float* B2  = (const float*)d_in[11];
  const float* G2  = (const float*)d_in[12];
  const float* H2  = (const float*)d_in[13];
  const float* W3  = (const float*)d_in[14];
  const float* B3  = (const float*)d_in[15];
  const float* G3  = (const float*)d_in[16];
  const float* H3  = (const float*)d_in[17];
  const float* W4  = (const float*)d_in[18];
  const float* B4  = (const float*)d_in[19];
  float* out = (float*)d_out;

  const int N = in_sizes[1];  // number of samples (indices count)

  int* wsI  = (int*)d_ws;
  int* cnt  = wsI;         // [64]
  int* cur  = wsI + 64;    // [64]
  int* off  = wsI + 128;   // [65]
  int* perm = wsI + 256;   // [N]

  zero_k<<<1, 128, 0, stream>>>(wsI);
  count_k<<<(N + 255) / 256, 256, 0, stream>>>(idx, cnt, N);
  scan_k<<<1, 64, 0, stream>>>(cnt, off);
  scatter_k<<<(N + 255) / 256, 256, 0, stream>>>(idx, off, cur, perm, N);
  mlp_k<<<NMODEL * BPM, 256, 0, stream>>>(x, perm, off,
                                          W0, B0, G0, H0,
                                          W1, B1, G1, H1,
                                          W2, B2, G2, H2,
                                          W3, B3, G3, H3,
                                          W4, B4, out);
}